// Earth4D_61229053771882
// MI455X (gfx1250) — compile-verified
//
#include <hip/hip_runtime.h>

typedef __attribute__((ext_vector_type(2))) float v2f;
typedef int v4i __attribute__((vector_size(16)));
typedef __attribute__((address_space(1))) v4i gv4i;   // global v4i
typedef __attribute__((address_space(3))) v4i lv4i;   // LDS v4i

#define TBL_LOG2 19
#define TBL      (1u << TBL_LOG2)
#define TMASK    (TBL - 1u)
#define HKY      2654435761u
#define HKZ      805459861u
#define NLEV     16

#if __has_builtin(__builtin_amdgcn_global_store_async_from_lds_b128)
#define USE_ASYNC_STORE 1
#define ASYNC_CPOL 1   /* TH[2:0]=1 -> th:TH_STORE_NT: write-once output, keep L2 for tables */
#else
#define USE_ASYNC_STORE 0
#endif

// One resolution level: gather 8 corners, trilinear blend, store result.
// TO_GLOBAL=true  -> NT store straight to global (fallback path)
// TO_GLOBAL=false -> plain store into LDS staging row (ds_store, compiler merges to b128)
template <bool TO_GLOBAL>
__device__ __forceinline__ void do_level(const v2f* __restrict__ tab, int l, bool dense,
                                         float X, float Y, float Z, v2f* __restrict__ o) {
    const v2f* t = tab + ((size_t)l << TBL_LOG2);
    float fr = (float)(32 << l);                  // res = floor(32 * 2^l), exact
    float px = X * fr, py = Y * fr, pz = Z * fr;
    float fx = floorf(px), fy = floorf(py), fz = floorf(pz);
    float wx = px - fx, wy = py - fy, wz = pz - fz;
    unsigned bx = (unsigned)(int)fx, by = (unsigned)(int)fy, bz = (unsigned)(int)fz;

    unsigned i0, i1, i2, i3, i4, i5, i6, i7;
    if (dense) {                                  // levels 0,1: (res+1)^3 <= 2^19
        unsigned s = (unsigned)(32 << l) + 1u, ss = s * s;
        unsigned b = bx + by * s + bz * ss;
        i0 = b;          i1 = b + 1u;
        i2 = b + s;      i3 = b + s + 1u;
        i4 = b + ss;     i5 = b + ss + 1u;
        i6 = b + ss + s; i7 = b + ss + s + 1u;
    } else {                                      // instant-ngp hash
        unsigned hy0 = by * HKY, hy1 = hy0 + HKY;
        unsigned hz0 = bz * HKZ, hz1 = hz0 + HKZ;
        unsigned e0 = hy0 ^ hz0, e1 = hy1 ^ hz0, e2 = hy0 ^ hz1, e3 = hy1 ^ hz1;
        unsigned x1 = bx + 1u;
        i0 = (bx ^ e0) & TMASK; i1 = (x1 ^ e0) & TMASK;
        i2 = (bx ^ e1) & TMASK; i3 = (x1 ^ e1) & TMASK;
        i4 = (bx ^ e2) & TMASK; i5 = (x1 ^ e2) & TMASK;
        i6 = (bx ^ e3) & TMASK; i7 = (x1 ^ e3) & TMASK;
    }

    // Issue all 8 gathers (global_load_b64) before consuming.
    v2f f0 = t[i0], f1 = t[i1], f2 = t[i2], f3 = t[i3];
    v2f f4 = t[i4], f5 = t[i5], f6 = t[i6], f7 = t[i7];

    float ux = 1.f - wx, uy = 1.f - wy, uz = 1.f - wz;
    float w00 = uy * uz, w10 = wy * uz, w01 = uy * wz, w11 = wy * wz;
    v2f acc = f0 * (ux * w00) + f1 * (wx * w00)
            + f2 * (ux * w10) + f3 * (wx * w10)
            + f4 * (ux * w01) + f5 * (wx * w01)
            + f6 * (ux * w11) + f7 * (wx * w11);

    if (TO_GLOBAL)
        __builtin_nontemporal_store(acc, o + l);
    else
        o[l] = acc;                               // LDS staging row
}

template <bool TO_GLOBAL>
__device__ __forceinline__ void encode3(const v2f* __restrict__ tab,
                                        float X, float Y, float Z,
                                        v2f* __restrict__ o) {
    // Software prefetch (global_prefetch_b8): warm the fully-random level-15
    // corner lines while levels 0..14 execute.
    {
        float fr = (float)(32 << (NLEV - 1));
        unsigned bx = (unsigned)(int)floorf(X * fr);
        unsigned by = (unsigned)(int)floorf(Y * fr);
        unsigned bz = (unsigned)(int)floorf(Z * fr);
        const v2f* t15 = tab + ((size_t)(NLEV - 1) << TBL_LOG2);
        unsigned hy0 = by * HKY, hy1 = hy0 + HKY;
        unsigned hz0 = bz * HKZ, hz1 = hz0 + HKZ;
        unsigned x1 = bx + 1u;
        __builtin_prefetch(t15 + ((bx ^ hy0 ^ hz0) & TMASK), 0, 3);
        __builtin_prefetch(t15 + ((x1 ^ hy0 ^ hz0) & TMASK), 0, 3);
        __builtin_prefetch(t15 + ((bx ^ hy1 ^ hz0) & TMASK), 0, 3);
        __builtin_prefetch(t15 + ((x1 ^ hy1 ^ hz0) & TMASK), 0, 3);
        __builtin_prefetch(t15 + ((bx ^ hy0 ^ hz1) & TMASK), 0, 3);
        __builtin_prefetch(t15 + ((x1 ^ hy0 ^ hz1) & TMASK), 0, 3);
        __builtin_prefetch(t15 + ((bx ^ hy1 ^ hz1) & TMASK), 0, 3);
        __builtin_prefetch(t15 + ((x1 ^ hy1 ^ hz1) & TMASK), 0, 3);
    }

    do_level<TO_GLOBAL>(tab, 0, true, X, Y, Z, o);
    do_level<TO_GLOBAL>(tab, 1, true, X, Y, Z, o);
#pragma unroll 2
    for (int l = 2; l < NLEV; ++l)
        do_level<TO_GLOBAL>(tab, l, false, X, Y, Z, o);
}

__device__ __forceinline__ void pick_enc(int enc, float4 p,
                                         const v2f* t0, const v2f* t1,
                                         const v2f* t2, const v2f* t3,
                                         const v2f*& tab, float& X, float& Y, float& Z) {
    if (enc == 0)      { tab = t0; X = p.x; Y = p.y; Z = p.z; }   // xyz
    else if (enc == 1) { tab = t1; X = p.x; Y = p.y; Z = p.w; }   // xyt
    else if (enc == 2) { tab = t2; X = p.y; Y = p.z; Z = p.w; }   // yzt
    else               { tab = t3; X = p.x; Y = p.z; Z = p.w; }   // xzt
}

// Grid: (ceil(N/256), 4). blockIdx.y selects the encoding -> 16384 wave32
// waves for latency hiding on the gather stream.
__global__ __launch_bounds__(256) void Earth4D_kernel(
    const float4* __restrict__ xyzt,
    const v2f* __restrict__ sp_tab,
    const v2f* __restrict__ xyt_tab,
    const v2f* __restrict__ yzt_tab,
    const v2f* __restrict__ xzt_tab,
    v2f* __restrict__ out, int n) {
    unsigned tid = threadIdx.x;
    int i = (int)(blockIdx.x * 256u + tid);
    int enc = blockIdx.y;

#if USE_ASYNC_STORE
    // 32 KB: [thread][level]; 128B-aligned so B128 DS/async accesses are legal.
    __shared__ __align__(128) v2f stage[256 * NLEV];
    bool full = ((int)((blockIdx.x + 1u) * 256u) <= n);
    if (full) {
        float4 p = xyzt[i];
        const v2f* tab; float X, Y, Z;
        pick_enc(enc, p, sp_tab, xyt_tab, yzt_tab, xzt_tab, tab, X, Y, Z);

        encode3<false>(tab, X, Y, Z, &stage[tid * NLEV]);

        // Wave-local: lanes only consume rows written by their own wave.
        asm volatile("s_wait_dscnt 0x0" ::: "memory");

        unsigned lane = tid & 31u;
        unsigned wb   = tid & ~31u;               // first thread of this wave
        unsigned j    = lane & 7u;                // 16B chunk within 128B row
        unsigned rsub = lane >> 3;                // 0..3
        char* gbase = (char*)out + ((size_t)blockIdx.x * 256u + wb) * 512
                    + (size_t)enc * 128 + (size_t)j * 16;
        char* lbase = (char*)stage + (size_t)wb * 128 + (size_t)j * 16;
#pragma unroll
        for (unsigned k = 0; k < 8; ++k) {
            unsigned r = k * 4u + rsub;           // row within wave, 0..31
            // LDS -> L2 direct, 4 fully-covered 128B lines per instruction, NT policy.
            __builtin_amdgcn_global_store_async_from_lds_b128(
                (gv4i*)(gbase + (size_t)r * 512),
                (lv4i*)(lbase + (size_t)r * 128), 0, ASYNC_CPOL);
        }
        asm volatile("s_wait_asynccnt 0x0" ::: "memory");
        return;
    }
#endif

    if (i >= n) return;
    float4 p = xyzt[i];
    const v2f* tab; float X, Y, Z;
    pick_enc(enc, p, sp_tab, xyt_tab, yzt_tab, xzt_tab, tab, X, Y, Z);
    v2f* o = out + (size_t)i * 64 + (size_t)enc * 16;
    encode3<true>(tab, X, Y, Z, o);
}

extern "C" void kernel_launch(void* const* d_in, const int* in_sizes, int n_in,
                              void* d_out, int out_size, void* d_ws, size_t ws_size,
                              hipStream_t stream) {
    const float4* xyzt   = (const float4*)d_in[0];
    const v2f*    sp_tab = (const v2f*)d_in[1];
    const v2f*    xyt_t  = (const v2f*)d_in[2];
    const v2f*    yzt_t  = (const v2f*)d_in[3];
    const v2f*    xzt_t  = (const v2f*)d_in[4];
    int n = in_sizes[0] / 4;

    dim3 block(256);
    dim3 grid((n + 255) / 256, 4);
    Earth4D_kernel<<<grid, block, 0, stream>>>(xyzt, sp_tab, xyt_t, yzt_t, xzt_t,
                                               (v2f*)d_out, n);
}